// GCNII_4166118277232
// MI455X (gfx1250) — compile-verified
//
#include <hip/hip_runtime.h>
#include <cmath>

typedef float v2f __attribute__((ext_vector_type(2)));
typedef float v8f __attribute__((ext_vector_type(8)));

#define HID 64

// ---------------- small elementwise kernels ----------------

__global__ void k_fill1(float* __restrict__ p, int n) {
    int i = blockIdx.x * blockDim.x + threadIdx.x;
    if (i < n) p[i] = 1.0f;   // self-loop contributes 1 to degree
}

__global__ void k_deg(const int* __restrict__ dst, float* __restrict__ deg, int E) {
    int i = blockIdx.x * blockDim.x + threadIdx.x;
    if (i < E) atomicAdd(&deg[dst[i]], 1.0f);
}

__global__ void k_rsqrt(float* __restrict__ p, int n) {
    int i = blockIdx.x * blockDim.x + threadIdx.x;
    if (i < n) p[i] = rsqrtf(p[i]);
}

// S[i,:] = dinv[i]^2 * Hin[i,:]   (self-loop term; also initializes S)
__global__ void k_self(const float* __restrict__ dinv, const float* __restrict__ Hin,
                       float* __restrict__ S, int n16) {
    int t = blockIdx.x * blockDim.x + threadIdx.x;
    if (t >= n16) return;
    int i = t >> 4;
    int c = (t & 15) << 2;
    float w = dinv[i]; w *= w;
    float4 v = *(const float4*)(Hin + (size_t)i * HID + c);
    float4 r;
    r.x = w * v.x; r.y = w * v.y; r.z = w * v.z; r.w = w * v.w;
    *(float4*)(S + (size_t)i * HID + c) = r;
}

// S[dst,:] += dinv[src]*dinv[dst] * Hin[src,:]   (16 lanes per edge, float4 each)
__global__ void k_spmm(const int* __restrict__ src, const int* __restrict__ dst,
                       const float* __restrict__ dinv, const float* __restrict__ Hin,
                       float* __restrict__ S, int e16) {
    int t = blockIdx.x * blockDim.x + threadIdx.x;
    if (t >= e16) return;
    int e = t >> 4;
    int c = (t & 15) << 2;
    int s = src[e];
    int d = dst[e];
    float w = dinv[s] * dinv[d];
    float4 v = *(const float4*)(Hin + (size_t)s * HID + c);
    float* o = S + (size_t)d * HID + c;
    atomicAdd(o + 0, w * v.x);
    atomicAdd(o + 1, w * v.y);
    atomicAdd(o + 2, w * v.z);
    atomicAdd(o + 3, w * v.w);
}

// S = (1-alpha)*S + alpha*H0   (initial residual)
__global__ void k_resid(float* __restrict__ S, const float* __restrict__ H0,
                        float alpha, int n16) {
    int t = blockIdx.x * blockDim.x + threadIdx.x;
    if (t >= n16) return;
    size_t off = ((size_t)(t >> 4)) * HID + ((t & 15) << 2);
    float4 s = *(const float4*)(S + off);
    float4 h = *(const float4*)(H0 + off);
    float ia = 1.0f - alpha;
    s.x = ia * s.x + alpha * h.x;
    s.y = ia * s.y + alpha * h.y;
    s.z = ia * s.z + alpha * h.z;
    s.w = ia * s.w + alpha * h.w;
    *(float4*)(S + off) = s;
}

// ---------------- fp32 WMMA GEMM: out[N,64] = epilogue(A[N,K] @ W[K,64]) ----------------
// One wave per 16x16 tile, 4 waves per block -> one block covers all 64 output cols
// of a 16-row stripe. Requires N % 16 == 0 (holds: 50000 = 3125*16) so EXEC is all-1s.
//
// Epilogues:
//   Sres != null : v = (1-beta)*Sres[m,n] + beta*acc      (GCNII identity mapping)
//   bias != null : v += bias[n]
//   relu         : v = max(v, 0)
__global__ void __launch_bounds__(128)
k_gemm64(const float* __restrict__ A, const float* __restrict__ W,
         const float* __restrict__ bias, const float* __restrict__ Sres,
         float beta, int relu, float* __restrict__ out, int K) {
    const int wave = threadIdx.x >> 5;
    const int lane = threadIdx.x & 31;
    const int half = lane >> 4;   // 0: lanes 0-15, 1: lanes 16-31
    const int lr   = lane & 15;
    const int r0 = blockIdx.x * 16;
    const int c0 = wave * 16;
    const float* Arow = A + (size_t)r0 * K;

    v8f acc = {0.f, 0.f, 0.f, 0.f, 0.f, 0.f, 0.f, 0.f};

#if defined(__has_builtin) && __has_builtin(__builtin_amdgcn_wmma_f32_16x16x4_f32)
    for (int k = 0; k < K; k += 4) {
        const int ka = k + half * 2;          // this half-wave's K pair
        // A fragment (16x4 f32): lane holds A[r0+lr][ka], A[r0+lr][ka+1]
        float2 at = *(const float2*)(Arow + (size_t)lr * K + ka);
        v2f a; a.x = at.x; a.y = at.y;
        // B fragment (4x16 f32): lane holds W[ka][c0+lr], W[ka+1][c0+lr]
        v2f b;
        b.x = W[(size_t)ka * HID + c0 + lr];
        b.y = W[(size_t)(ka + 1) * HID + c0 + lr];
        acc = __builtin_amdgcn_wmma_f32_16x16x4_f32(
            /*neg_a=*/false, a, /*neg_b=*/false, b,
            /*c_mod=*/(short)0, acc, /*reuse_a=*/false, /*reuse_b=*/false);
    }
#else
    // Scalar fallback with the same C/D output mapping (compile hedge only).
    for (int k = 0; k < K; ++k) {
        float bv = W[(size_t)k * HID + c0 + lr];
#pragma unroll
        for (int i = 0; i < 8; ++i)
            acc[i] += Arow[(size_t)(i + half * 8) * K + k] * bv;
    }
#endif

    // C/D layout: VGPR i -> row M = i (lanes 0-15) or i+8 (lanes 16-31), col N = lane%16
#pragma unroll
    for (int i = 0; i < 8; ++i) {
        int m = r0 + i + half * 8;
        int n = c0 + lr;
        float v = acc[i];
        if (Sres) v = (1.0f - beta) * Sres[(size_t)m * HID + n] + beta * v;
        if (bias) v += bias[n];
        if (relu) v = fmaxf(v, 0.0f);
        out[(size_t)m * HID + n] = v;
    }
}

// ---------------- host orchestration ----------------

extern "C" void kernel_launch(void* const* d_in, const int* in_sizes, int n_in,
                              void* d_out, int out_size, void* d_ws, size_t ws_size,
                              hipStream_t stream) {
    const float* feature = (const float*)d_in[0];
    const float* W0      = (const float*)d_in[1];
    const float* b0      = (const float*)d_in[2];
    const float* Wc      = (const float*)d_in[3];
    const float* Wl      = (const float*)d_in[4];
    const float* bl      = (const float*)d_in[5];
    const int*   src     = (const int*)d_in[6];
    const int*   dst     = (const int*)d_in[7];
    float* out = (float*)d_out;

    const int IN = 512;
    const int N  = in_sizes[0] / IN;            // 50000
    const int E  = in_sizes[6];                 // 1600000
    const int L  = in_sizes[3] / (HID * HID);   // 32
    const float ALPHA = 0.1f, LAMBDA = 0.5f;

    // workspace layout: dinv[N] | H0[N*64] | H[N*64] | S[N*64]  (~38.6 MB)
    float* dinv = (float*)d_ws;
    float* H0   = dinv + (((size_t)N + 63) & ~(size_t)63);
    float* Hc   = H0 + (size_t)N * HID;
    float* S    = Hc + (size_t)N * HID;

    const int n16 = N * 16;          // 800000
    const int e16 = E * 16;          // 25.6M

    // D^{-1/2} with self loops
    k_fill1 <<<(N + 255) / 256, 256, 0, stream>>>(dinv, N);
    k_deg   <<<(E + 255) / 256, 256, 0, stream>>>(dst, dinv, E);
    k_rsqrt <<<(N + 255) / 256, 256, 0, stream>>>(dinv, N);

    // H0 = relu(feature @ W0 + b0)
    k_gemm64<<<N / 16, 128, 0, stream>>>(feature, W0, b0, nullptr, 0.0f, 1, H0, IN);

    for (int l = 0; l < L; ++l) {
        const float* Hin = (l == 0) ? H0 : Hc;
        float beta = logf(LAMBDA / (float)(l + 1) + 1.0f);

        // S = A_norm @ Hin
        k_self <<<(n16 + 255) / 256, 256, 0, stream>>>(dinv, Hin, S, n16);
        k_spmm <<<(e16 + 255) / 256, 256, 0, stream>>>(src, dst, dinv, Hin, S, e16);
        // S = (1-alpha)*S + alpha*H0
        k_resid<<<(n16 + 255) / 256, 256, 0, stream>>>(S, H0, ALPHA, n16);
        // Hc = relu((1-beta)*S + beta*(S @ Wc[l]))
        k_gemm64<<<N / 16, 128, 0, stream>>>(S, Wc + (size_t)l * HID * HID,
                                             nullptr, S, beta, 1, Hc, HID);
    }

    // out = Hc @ Wl + bl
    k_gemm64<<<N / 16, 128, 0, stream>>>(Hc, Wl, bl, nullptr, 0.0f, 0, out, HID);
}